// VGATEncoder_8564164788730
// MI455X (gfx1250) — compile-verified
//
#include <hip/hip_runtime.h>
#include <hip/hip_bf16.h>

typedef __attribute__((ext_vector_type(2))) float v2f;
typedef __attribute__((ext_vector_type(8))) float v8f;

#define NNODES 50000
#define NEDGES 800000
#define ETOT   (NNODES + NEDGES)
#define INC    128
#define HIDC   64
#define NHEADS 2
#define LATC   32
#define NEG_SLOPE 0.2f

// ---------------------------------------------------------------------------
// Order-preserving float <-> uint key for atomicMax-based segment max.
// ---------------------------------------------------------------------------
__device__ __forceinline__ unsigned fkey(float f) {
  unsigned u = __float_as_uint(f);
  return (u & 0x80000000u) ? ~u : (u | 0x80000000u);
}
__device__ __forceinline__ float funkey(unsigned u) {
  u = (u & 0x80000000u) ? (u & 0x7FFFFFFFu) : ~u;
  return __uint_as_float(u);
}

// Edge e: for e < NEDGES use edge_index rows [src | dst]; else self-loop.
__device__ __forceinline__ void edge_sd(const int* __restrict__ ei, int e,
                                        int& s, int& d) {
  if (e < NEDGES) { s = ei[e]; d = ei[NEDGES + e]; }
  else            { s = d = e - NEDGES; }
}

__device__ __forceinline__ float lrelu(float x) {
  return x > 0.f ? x : NEG_SLOPE * x;
}

// ---------------------------------------------------------------------------
// FP32 WMMA GEMM:  C[nrows x M] = A[nrows x K] * W[K x M]
// One wave computes 16 rows x M cols via M/16 tiles of v_wmma_f32_16x16x4_f32.
// ISA 7.12.2 layout (32-bit A 16x4): lane L holds A[m = L%16][k0+v],
// k0 = (L/16)*2, v = frag element; B mirrored (n = L%16, same k striping);
// C/D: VGPR r, lane L -> row r + (L/16)*8, col L%16.
// nrows must be a multiple of 16 (50000 is).
// ---------------------------------------------------------------------------
template <int K, int M>
__global__ __launch_bounds__(256)
void gemm_wmma_f32_k(const float* __restrict__ A, const float* __restrict__ W,
                     float* __restrict__ C, int nrows) {
  const int lane = threadIdx.x & 31;
  const int wave = blockIdx.x * (blockDim.x >> 5) + (threadIdx.x >> 5);
  const int rowBase = wave * 16;
  if (rowBase >= nrows) return;           // wave-uniform: EXEC stays all-ones
  const int half = lane >> 4;             // 0: K lo pair, 1: K hi pair
  const int mn   = lane & 15;             // row (A) / col (B, C)
  const int koff = half * 2;
  constexpr int NT = M / 16;

  v8f acc[NT] = {};
  const float* arow = A + (size_t)(rowBase + mn) * K;

  for (int kk = 0; kk < K; kk += 4) {
    v2f a;
    a.x = arow[kk + koff];
    a.y = arow[kk + koff + 1];
#pragma unroll
    for (int t = 0; t < NT; ++t) {
      v2f b;
      b.x = W[(size_t)(kk + koff)     * M + t * 16 + mn];
      b.y = W[(size_t)(kk + koff + 1) * M + t * 16 + mn];
      acc[t] = __builtin_amdgcn_wmma_f32_16x16x4_f32(
          /*neg_a=*/false, a, /*neg_b=*/false, b,
          /*c_mod=*/(short)0, acc[t], /*reuse_a=*/false, /*reuse_b=*/false);
    }
  }
#pragma unroll
  for (int t = 0; t < NT; ++t)
#pragma unroll
    for (int r = 0; r < 8; ++r)
      C[(size_t)(rowBase + r + half * 8) * M + t * 16 + mn] = acc[t][r];
}

// ---------------------------------------------------------------------------
// Per-node attention scalars: a_src[n,h] = dot(h[n,h,:], att_src[h,:]), ditto dst.
// ---------------------------------------------------------------------------
template <int H, int HC>
__global__ void att_dots_k(const float* __restrict__ h,
                           const float* __restrict__ att_s,
                           const float* __restrict__ att_d,
                           float* __restrict__ a_s, float* __restrict__ a_d) {
  int i = blockIdx.x * blockDim.x + threadIdx.x;   // node*H + head
  if (i >= NNODES * H) return;
  int node = i / H, hd = i % H;
  const float* row = h + (size_t)node * (H * HC) + hd * HC;
  float ss = 0.f, sd = 0.f;
#pragma unroll
  for (int c = 0; c < HC; ++c) {
    float v = row[c];
    ss += v * att_s[hd * HC + c];
    sd += v * att_d[hd * HC + c];
  }
  a_s[i] = ss;
  a_d[i] = sd;
}

// ---------------------------------------------------------------------------
// Pass A: segment max of leaky-relu logits per destination node.
// ---------------------------------------------------------------------------
template <int H>
__global__ void edge_max_k(const int* __restrict__ ei,
                           const float* __restrict__ a_s,
                           const float* __restrict__ a_d,
                           unsigned* __restrict__ nodemax) {
  int i = blockIdx.x * blockDim.x + threadIdx.x;
  if (i >= ETOT * H) return;
  int e = i / H, hd = i % H;
  int s, d; edge_sd(ei, e, s, d);
  float logit = lrelu(a_s[s * H + hd] + a_d[d * H + hd]);
  atomicMax(&nodemax[d * H + hd], fkey(logit));
}

// ---------------------------------------------------------------------------
// Pass B: ealpha = exp(logit - max[dst]); denom[dst] += ealpha.
// ---------------------------------------------------------------------------
template <int H>
__global__ void edge_exp_k(const int* __restrict__ ei,
                           const float* __restrict__ a_s,
                           const float* __restrict__ a_d,
                           const unsigned* __restrict__ nodemax,
                           float* __restrict__ ealpha,
                           float* __restrict__ denom) {
  int i = blockIdx.x * blockDim.x + threadIdx.x;
  if (i >= ETOT * H) return;
  int e = i / H, hd = i % H;
  int s, d; edge_sd(ei, e, s, d);
  float logit = lrelu(a_s[s * H + hd] + a_d[d * H + hd]);
  float ea = __expf(logit - funkey(nodemax[d * H + hd]));
  ealpha[i] = ea;
  atomicAdd(&denom[d * H + hd], ea);
}

// ---------------------------------------------------------------------------
// Pass C: out[dst, c] += (ealpha/denom) * h[src, c].  256 threads cover
// (256/C) edges x C channels.
// ---------------------------------------------------------------------------
template <int H, int C>
__global__ __launch_bounds__(256)
void edge_agg_k(const int* __restrict__ ei,
                const float* __restrict__ ealpha,
                const float* __restrict__ denom,
                const float* __restrict__ hsrc,
                float* __restrict__ out) {
  constexpr int EPB = 256 / C;
  int t = threadIdx.x;
  int e = blockIdx.x * EPB + t / C;
  if (e >= ETOT) return;
  int c  = t % C;
  int hd = c / (C / H);
  int s, d; edge_sd(ei, e, s, d);
  float al = ealpha[e * H + hd] / (denom[d * H + hd] + 1e-16f);
  atomicAdd(&out[(size_t)d * C + c], al * hsrc[(size_t)s * C + c]);
}

// ---------------------------------------------------------------------------
// Epilogues.
// ---------------------------------------------------------------------------
__global__ void bias_elu_k(float* __restrict__ h, const float* __restrict__ b,
                           int n, int m) {
  int i = blockIdx.x * blockDim.x + threadIdx.x;
  if (i >= n) return;
  float v = h[i] + b[i % m];
  h[i] = v > 0.f ? v : expm1f(v);
}

__global__ void bias_add_k(float* __restrict__ o, const float* __restrict__ b,
                           int n, int m) {
  int i = blockIdx.x * blockDim.x + threadIdx.x;
  if (i >= n) return;
  o[i] += b[i % m];
}

// ---------------------------------------------------------------------------
// Orchestration.
// ---------------------------------------------------------------------------
static inline int cdiv(long long a, long long b) { return (int)((a + b - 1) / b); }

extern "C" void kernel_launch(void* const* d_in, const int* in_sizes, int n_in,
                              void* d_out, int out_size, void* d_ws, size_t ws_size,
                              hipStream_t stream) {
  const float* x    = (const float*)d_in[0];
  const int*   ei   = (const int*)d_in[1];
  const float* W1   = (const float*)d_in[2];
  const float* as1  = (const float*)d_in[3];
  const float* ad1  = (const float*)d_in[4];
  const float* b1   = (const float*)d_in[5];
  const float* Wmu  = (const float*)d_in[6];
  const float* asmu = (const float*)d_in[7];
  const float* admu = (const float*)d_in[8];
  const float* bmu  = (const float*)d_in[9];
  const float* Wlv  = (const float*)d_in[10];
  const float* aslv = (const float*)d_in[11];
  const float* adlv = (const float*)d_in[12];
  const float* blv  = (const float*)d_in[13];

  float* out_mu = (float*)d_out;
  float* out_lv = out_mu + (size_t)NNODES * LATC;

  // Workspace carve-out (~72 MB).
  float* ws = (float*)d_ws;
  float* h1 = ws;  ws += (size_t)NNODES * INC;          // x @ W1
  float* h2 = ws;  ws += (size_t)NNODES * INC;          // aggregated + ELU
  float* a_s = ws; ws += (size_t)NNODES * NHEADS;
  float* a_d = ws; ws += (size_t)NNODES * NHEADS;
  unsigned* nmax = (unsigned*)ws; ws += (size_t)NNODES * NHEADS;
  float* den = ws; ws += (size_t)NNODES * NHEADS;
  float* eal = ws; ws += (size_t)ETOT * NHEADS;
  float* hmu = ws; ws += (size_t)NNODES * LATC;
  float* hlv = ws; ws += (size_t)NNODES * LATC;
  // Layer-2 softmax scratch reuses a_s/a_d/nmax/den/eal (H=1 fits in H=2 space).

  const int B = 256;
  const int gemm_blocks = cdiv(NNODES / 16, 8);         // 8 waves/block

  // ---------------- Layer 1: GAT(128 -> 2 heads x 64, concat) ----------------
  gemm_wmma_f32_k<INC, INC><<<gemm_blocks, B, 0, stream>>>(x, W1, h1, NNODES);
  att_dots_k<NHEADS, HIDC><<<cdiv((size_t)NNODES * NHEADS, B), B, 0, stream>>>(
      h1, as1, ad1, a_s, a_d);
  hipMemsetAsync(nmax, 0, (size_t)NNODES * NHEADS * 4, stream);
  hipMemsetAsync(den,  0, (size_t)NNODES * NHEADS * 4, stream);
  hipMemsetAsync(h2,   0, (size_t)NNODES * INC * 4, stream);
  edge_max_k<NHEADS><<<cdiv((size_t)ETOT * NHEADS, B), B, 0, stream>>>(ei, a_s, a_d, nmax);
  edge_exp_k<NHEADS><<<cdiv((size_t)ETOT * NHEADS, B), B, 0, stream>>>(
      ei, a_s, a_d, nmax, eal, den);
  edge_agg_k<NHEADS, INC><<<cdiv(ETOT, 256 / INC), B, 0, stream>>>(ei, eal, den, h1, h2);
  bias_elu_k<<<cdiv((size_t)NNODES * INC, B), B, 0, stream>>>(h2, b1, NNODES * INC, INC);

  // ---------------- Layer 2a: mu = GAT(128 -> 1 head x 32) -------------------
  gemm_wmma_f32_k<INC, LATC><<<gemm_blocks, B, 0, stream>>>(h2, Wmu, hmu, NNODES);
  att_dots_k<1, LATC><<<cdiv(NNODES, B), B, 0, stream>>>(hmu, asmu, admu, a_s, a_d);
  hipMemsetAsync(nmax, 0, (size_t)NNODES * 4, stream);
  hipMemsetAsync(den,  0, (size_t)NNODES * 4, stream);
  hipMemsetAsync(out_mu, 0, (size_t)NNODES * LATC * 4, stream);
  edge_max_k<1><<<cdiv(ETOT, B), B, 0, stream>>>(ei, a_s, a_d, nmax);
  edge_exp_k<1><<<cdiv(ETOT, B), B, 0, stream>>>(ei, a_s, a_d, nmax, eal, den);
  edge_agg_k<1, LATC><<<cdiv(ETOT, 256 / LATC), B, 0, stream>>>(ei, eal, den, hmu, out_mu);
  bias_add_k<<<cdiv((size_t)NNODES * LATC, B), B, 0, stream>>>(out_mu, bmu, NNODES * LATC, LATC);

  // ---------------- Layer 2b: logvar = GAT(128 -> 1 head x 32) ---------------
  gemm_wmma_f32_k<INC, LATC><<<gemm_blocks, B, 0, stream>>>(h2, Wlv, hlv, NNODES);
  att_dots_k<1, LATC><<<cdiv(NNODES, B), B, 0, stream>>>(hlv, aslv, adlv, a_s, a_d);
  hipMemsetAsync(nmax, 0, (size_t)NNODES * 4, stream);
  hipMemsetAsync(den,  0, (size_t)NNODES * 4, stream);
  hipMemsetAsync(out_lv, 0, (size_t)NNODES * LATC * 4, stream);
  edge_max_k<1><<<cdiv(ETOT, B), B, 0, stream>>>(ei, a_s, a_d, nmax);
  edge_exp_k<1><<<cdiv(ETOT, B), B, 0, stream>>>(ei, a_s, a_d, nmax, eal, den);
  edge_agg_k<1, LATC><<<cdiv(ETOT, 256 / LATC), B, 0, stream>>>(ei, eal, den, hlv, out_lv);
  bias_add_k<<<cdiv((size_t)NNODES * LATC, B), B, 0, stream>>>(out_lv, blv, NNODES * LATC, LATC);
}